// Model_33268816675182
// MI455X (gfx1250) — compile-verified
//
#include <hip/hip_runtime.h>
#include <stdint.h>

// Problem constants (from reference)
#define Bq   4
#define Cq   64
#define Hq   128
#define Wq   512
#define SWq  512
#define HALF 256
#define HWq  (Hq * Wq)          // 65536 = 2^16
#define NPIX (Bq * HWq)         // 262144
#define CELLS_PER_B (SWq * SWq) // 262144 = 2^18
#define NSEG (Bq * CELLS_PER_B) // 1048576

#define CELLS_PER_BLOCK 1024    // 256 threads * 4 cells
#define WRITE_BLOCKS (NSEG / CELLS_PER_BLOCK) // 1024

typedef float v4f __attribute__((ext_vector_type(4)));
typedef unsigned long long u64;
typedef unsigned long long v2u __attribute__((ext_vector_type(2)));
typedef unsigned int u32x4 __attribute__((ext_vector_type(4)));
typedef int i32x8 __attribute__((ext_vector_type(8)));
typedef int i32x4 __attribute__((ext_vector_type(4)));

#if __has_builtin(__builtin_amdgcn_tensor_load_to_lds) && \
    __has_builtin(__builtin_amdgcn_s_wait_tensorcnt)
#define USE_TDM 1
#else
#define USE_TDM 0
#endif

// Monotonic float->uint mapping: f0 < f1  <=>  ord(f0) < ord(f1) (unsigned)
__device__ __forceinline__ unsigned int order_f32(float f) {
    unsigned int b = __float_as_uint(f);
    return (b & 0x80000000u) ? ~b : (b | 0x80000000u);
}

// ---------------------------------------------------------------------------
// Pass 0: sentinel-fill the per-cell key array (8 MB, stays hot in L2)
// ---------------------------------------------------------------------------
__global__ void init_keys(u64* __restrict__ keys) {
    int t = blockIdx.x * blockDim.x + threadIdx.x; // NSEG/2 threads
    v2u s; s.x = ~0ull; s.y = ~0ull;
    *(v2u*)(keys + 2 * t) = s;                     // b128 store
}

// ---------------------------------------------------------------------------
// Pass 1: project every pixel, atomic-min a packed {ordered(y), pixel} key.
// Winner per BEV cell = min y (== argmax of max_h - y); ties -> lowest pixel
// id, so the result is deterministic across graph replays.
// ---------------------------------------------------------------------------
__global__ void scatter_min(const float* __restrict__ camK,
                            const float* __restrict__ depth,
                            const float* __restrict__ mpp_p,
                            u64* __restrict__ keys) {
    int p = blockIdx.x * blockDim.x + threadIdx.x; // 0 .. NPIX-1
    int b = p >> 16;                               // / HWq (uniform per block)
    int r = p & 0xFFFF;
    int v = r >> 9;                                // / Wq
    int u = r & (Wq - 1);

    // Row-scale K: scale = [[W/ORI_W],[H/ORI_H],[1]] = [[0.5],[0.5],[1]]
    const float sx = (float)Wq / 1024.0f;
    const float sy = (float)Hq / 256.0f;
    const float* K = camK + b * 9;
    float a  = K[0] * sx, bb = K[1] * sx, c  = K[2] * sx;
    float d  = K[3] * sy, e  = K[4] * sy, f  = K[5] * sy;
    float g  = K[6],      h  = K[7],      ii = K[8];

    // General 3x3 inverse via adjugate (matches jnp.linalg.inv semantics)
    float A  = e * ii - f * h;
    float Bm = f * g  - d * ii;
    float Cm = d * h  - e * g;
    float rdet = 1.0f / (a * A + bb * Bm + c * Cm);
    float i00 = A * rdet,  i01 = (c * h - bb * ii) * rdet, i02 = (bb * f - c * e) * rdet;
    float i10 = Bm * rdet, i11 = (a * ii - c * g) * rdet,  i12 = (c * d - a * f) * rdet;
    float i20 = Cm * rdet, i21 = (bb * g - a * h) * rdet,  i22 = (a * e - bb * d) * rdet;

    float uf = (float)u, vf = (float)v;
    float dep = depth[p] * 1.2f;
    float X = (i00 * uf + i01 * vf + i02) * dep;
    float Y = (i10 * uf + i11 * vf + i12) * dep;
    float Z = (i20 * uf + i21 * vf + i22) * dep;

    float mpp = mpp_p[0];
    float x = truncf(X / mpp);
    float z = truncf(Z / mpp);

    bool kept = (x >= -(float)HALF) && (x <= (float)(HALF - 1)) &&
                (z >= -(float)HALF) && (z <= (float)(HALF - 1));
    if (kept) {
        int xi = (int)x + HALF;
        int zi = (int)z + HALF;
        int cell = (b << 18) + (xi << 9) + zi;
        u64 packed = ((u64)order_f32(Y) << 32) | (unsigned int)p;
        atomicMin(&keys[cell], packed); // global_atomic_min_u64
    }
}

// ---------------------------------------------------------------------------
// Pass 2: each block owns 1024 consecutive cells. The block's 8 KB key slice
// is DMA'd into LDS by the Tensor Data Mover (1-D tile, data_size=8B),
// tracked with TENSORcnt; then one thread per 4 consecutive zi cells does
// predicated winner-feature gathers (image fits in the 192 MB L2) and fully
// coalesced 16 B non-temporal stores into the 268 MB output stream.
// ---------------------------------------------------------------------------
__global__ void write_out(const float* __restrict__ img,
                          const u64* __restrict__ keys,
                          float* __restrict__ out) {
    __shared__ u64 skeys[CELLS_PER_BLOCK];         // 8 KB

    int t = blockIdx.x * blockDim.x + threadIdx.x; // 0 .. NSEG/4-1
    int cell0 = t << 2;                            // 4 consecutive zi
    int b   = cell0 >> 18;                         // uniform per block
    int rem = cell0 & (CELLS_PER_B - 1);           // xi*512 + zi0

    u64 k0, k1, k2, k3;
#if USE_TDM
    {
        if (threadIdx.x < 32) {
            // Tensor DMA descriptor (D#), packed per CDNA5 ISA §8.3/8.4
            unsigned lds_off = (unsigned)(uintptr_t)&skeys[0]; // flat[31:0] == LDS byte offset
            u64 gaddr = (u64)(uintptr_t)(keys + (size_t)blockIdx.x * CELLS_PER_BLOCK);

            u32x4 g0;
            g0.x = 1u;                                   // count=1, is_restore=0, gather off
            g0.y = lds_off;                              // lds_addr (bytes)
            g0.z = (unsigned)(gaddr & 0xFFFFFFFFull);    // global_addr[31:0]
            g0.w = (unsigned)((gaddr >> 32) & 0x1FFFFFFull) | (2u << 30); // addr[56:32], type=2

            i32x8 g1;
            g1[0] = (int)(3u << 16);          // workgroup_mask=0, data_size=3 (8 B)
            g1[1] = (int)(CELLS_PER_BLOCK << 16); // tensor_dim0[15:0] in bits[63:48]
            g1[2] = (int)(1u << 16);          // tensor_dim0[31:16]=0, tensor_dim1[15:0]=1
            g1[3] = (int)(CELLS_PER_BLOCK << 16); // tensor_dim1[31:16]=0, tile_dim0=1024
            g1[4] = 0;                        // tile_dim1=0 (unused), tile_dim2=0 (1-D tile)
            g1[5] = (int)CELLS_PER_BLOCK;     // tensor_dim0_stride[31:0]
            g1[6] = 0;                        // stride0[47:32]=0, stride1[15:0]=0
            g1[7] = 0;                        // stride1[47:16]=0

            i32x4 gz4 = {0, 0, 0, 0};         // groups 2/3 unused (<=2-D tensor)
            i32x8 gz8 = {0, 0, 0, 0, 0, 0, 0, 0};
            // 6-arg form (this toolchain): (g0, g1, g2, g3, extra_group, cpol)
            __builtin_amdgcn_tensor_load_to_lds(g0, g1, gz4, gz4, gz8, 0);
            __builtin_amdgcn_s_wait_tensorcnt(0);
        }
        __syncthreads();
        const u64* kp = &skeys[(threadIdx.x << 2)];
        k0 = kp[0]; k1 = kp[1]; k2 = kp[2]; k3 = kp[3]; // ds_load_b128 x2
    }
#else
    {
        const u64* kp = keys + cell0;                   // coalesced fallback
        k0 = kp[0]; k1 = kp[1]; k2 = kp[2]; k3 = kp[3];
    }
#endif

    bool h0 = (k0 != ~0ull), h1 = (k1 != ~0ull), h2 = (k2 != ~0ull), h3 = (k3 != ~0ull);

    // Winner pixel p (low 32 bits): image offset = (p>>16)*Cq*HWq + (p&0xFFFF) + c*HWq
    const float* base0 = img + ((k0 & 0xFFFFFFFFull) >> 16) * (unsigned)(Cq * HWq)
                             + (unsigned)(k0 & 0xFFFFull);
    const float* base1 = img + ((k1 & 0xFFFFFFFFull) >> 16) * (unsigned)(Cq * HWq)
                             + (unsigned)(k1 & 0xFFFFull);
    const float* base2 = img + ((k2 & 0xFFFFFFFFull) >> 16) * (unsigned)(Cq * HWq)
                             + (unsigned)(k2 & 0xFFFFull);
    const float* base3 = img + ((k3 & 0xFFFFFFFFull) >> 16) * (unsigned)(Cq * HWq)
                             + (unsigned)(k3 & 0xFFFFull);

    if (h0) __builtin_prefetch(base0, 0, 3);            // global_prefetch_b8
    if (h1) __builtin_prefetch(base1, 0, 3);
    if (h2) __builtin_prefetch(base2, 0, 3);
    if (h3) __builtin_prefetch(base3, 0, 3);

    float* op = out + (size_t)b * (Cq * CELLS_PER_B) + rem;

#pragma unroll 4
    for (int c = 0; c < Cq; ++c) {
        v4f val;
        val.x = h0 ? base0[c * HWq] : 0.0f;
        val.y = h1 ? base1[c * HWq] : 0.0f;
        val.z = h2 ? base2[c * HWq] : 0.0f;
        val.w = h3 ? base3[c * HWq] : 0.0f;
        __builtin_nontemporal_store(val, (v4f*)(op + (size_t)c * CELLS_PER_B));
    }
}

// ---------------------------------------------------------------------------
extern "C" void kernel_launch(void* const* d_in, const int* in_sizes, int n_in,
                              void* d_out, int out_size, void* d_ws, size_t ws_size,
                              hipStream_t stream) {
    const float* img   = (const float*)d_in[0]; // (B,C,H,W)
    const float* camK  = (const float*)d_in[1]; // (B,3,3)
    const float* depth = (const float*)d_in[2]; // (B,H,W)
    const float* mpp   = (const float*)d_in[3]; // scalar
    float* out = (float*)d_out;                 // (B,C,512,512)
    u64* keys = (u64*)d_ws;                     // NSEG u64 = 8 MB

    init_keys  <<<NSEG / 512, 256, 0, stream>>>(keys);
    scatter_min<<<NPIX / 256, 256, 0, stream>>>(camK, depth, mpp, keys);
    write_out  <<<WRITE_BLOCKS, 256, 0, stream>>>(img, keys, out);
}